// GNN_33586644254844
// MI455X (gfx1250) — compile-verified
//
#include <hip/hip_runtime.h>
#include <hip/hip_bf16.h>
#include <math.h>

#define NN 256
#define PP (NN * (NN - 1) / 2)   // 32640 pairs
#define BB 32
#define HH 64
#define HS 65                    // padded LDS row stride (odd -> no bank conflicts)

typedef __attribute__((ext_vector_type(2))) float v2f;
typedef __attribute__((ext_vector_type(8))) float v8f;

// ---------------------------------------------------------------------------
// Kernel A (bandwidth-bound): per batch, collapse the graph to 6 scalars.
//   deg[n]  = #incident pairs with dec==1  (only per-node quantity needed)
//   S[b]    = { sum(deg)/255, #(deg==0), 2.0 }   (column sums of node_feat)
//   EF[b]   = { dec[cur]==1, dec[cur]!=0.5, 1 }  (edge features of selected edge)
// ---------------------------------------------------------------------------
__global__ void gnn_feat_kernel(const float* __restrict__ x, float* __restrict__ ws) {
  const int b = blockIdx.x;
  const int n = threadIdx.x;                 // node id 0..255
  const float* dec = x + (size_t)b * (2 * PP);
  const float* ind = dec + PP;

  // deg[n]: walk all 255 partners; pair index p(i<j) = i*(2N-1-i)/2 + (j-i-1)
  int deg = 0;
  #pragma unroll 4
  for (int m = 0; m < NN; ++m) {
    if (m == n) continue;
    const int i = (n < m) ? n : m;
    const int j = (n < m) ? m : n;
    const int p = ((i * (2 * NN - 1 - i)) >> 1) + (j - i - 1);
    deg += (dec[p] == 1.0f) ? 1 : 0;
  }

  __shared__ int sdeg[NN];
  __shared__ int scur;
  sdeg[n] = deg;
  if (n == 0) scur = 0;
  __syncthreads();

  // ind is exactly one-hot: exactly one thread writes scur (race-free)
  for (int k = n; k < PP; k += NN) {
    if (ind[k] > 0.5f) scur = k;
  }
  __syncthreads();

  if (n == 0) {
    int sum = 0, iso = 0;
    for (int t = 0; t < NN; ++t) { sum += sdeg[t]; iso += (sdeg[t] == 0) ? 1 : 0; }
    const float d = dec[scur];
    float* S  = ws + b * 3;
    float* EF = ws + BB * 3 + b * 3;
    S[0] = (float)sum * (1.0f / 255.0f);     // sum(deg/(N-1))
    S[1] = (float)iso;                       // sum(deg==0)
    S[2] = 2.0f;                             // sum(attached) (i!=j always)
    EF[0] = (d == 1.0f) ? 1.0f : 0.0f;
    EF[1] = (d != 0.5f) ? 1.0f : 0.0f;
    EF[2] = 1.0f;                            // ind[cur] == 1
  }
}

// ---------------------------------------------------------------------------
// WMMA helpers: D = relu(A(32x64) @ W(64x64) + bias), via v_wmma_f32_16x16x4_f32.
// Each wave owns one 16x16 output tile (mt in {0,1}, nt in {0..3}); K=64 -> 16
// chained WMMA steps. VGPR layouts per CDNA5 ISA §7.12.2:
//   A: lane = m + 16*(k>=2), vgpr = k&1     C/D: vgpr r -> m = r + 8*(lane>=16), n = lane&15
//   B assumed transpose-symmetric to A: lane = n + 16*(k>=2), vgpr = k&1
// ---------------------------------------------------------------------------
__device__ __forceinline__ void wmma_tile_layer(const float* __restrict__ Hin,
                                                float* __restrict__ Hout,
                                                const float* __restrict__ Wg,
                                                const float* __restrict__ bias,
                                                int lane, int mt, int nt) {
  const int ml = lane & 15;
  const int hi = lane >> 4;                  // 0 or 1
  const int mg = 16 * mt + ml;               // A row
  const int ng = 16 * nt + ml;               // B/D column
  v8f acc = {};
  #pragma unroll
  for (int step = 0; step < 16; ++step) {
    const int k0 = 4 * step + 2 * hi;
    v2f a, bf;
    a.x  = Hin[mg * HS + k0];
    a.y  = Hin[mg * HS + k0 + 1];
    bf.x = Wg[k0 * HH + ng];
    bf.y = Wg[(k0 + 1) * HH + ng];
    acc = __builtin_amdgcn_wmma_f32_16x16x4_f32(false, a, false, bf,
                                                (short)0, acc, false, false);
  }
  const float bn = bias[ng];
  #pragma unroll
  for (int r = 0; r < 8; ++r) {
    const int mrow = 16 * mt + r + 8 * hi;
    Hout[mrow * HS + ng] = fmaxf(acc[r] + bn, 0.0f);
  }
}

// Edge-head layer: W = We1[0:64] + We1[64:128] (src/dst halves folded since
// h_src == h_dst), epilogue adds be1 + EF(b)·We1[128:131].
__device__ __forceinline__ void wmma_edge_layer(const float* __restrict__ Hin,
                                                float* __restrict__ Hout,
                                                const float* __restrict__ We1,
                                                const float* __restrict__ be1,
                                                const float* __restrict__ EF,
                                                int lane, int mt, int nt) {
  const int ml = lane & 15;
  const int hi = lane >> 4;
  const int mg = 16 * mt + ml;
  const int ng = 16 * nt + ml;
  v8f acc = {};
  #pragma unroll
  for (int step = 0; step < 16; ++step) {
    const int k0 = 4 * step + 2 * hi;
    v2f a, bf;
    a.x  = Hin[mg * HS + k0];
    a.y  = Hin[mg * HS + k0 + 1];
    bf.x = We1[k0 * HH + ng]       + We1[(k0 + HH) * HH + ng];
    bf.y = We1[(k0 + 1) * HH + ng] + We1[(k0 + 1 + HH) * HH + ng];
    acc = __builtin_amdgcn_wmma_f32_16x16x4_f32(false, a, false, bf,
                                                (short)0, acc, false, false);
  }
  const float bn = be1[ng];
  const float w0 = We1[(2 * HH + 0) * HH + ng];
  const float w1 = We1[(2 * HH + 1) * HH + ng];
  const float w2 = We1[(2 * HH + 2) * HH + ng];
  #pragma unroll
  for (int r = 0; r < 8; ++r) {
    const int mrow = 16 * mt + r + 8 * hi;
    const float e = EF[mrow * 3 + 0] * w0 + EF[mrow * 3 + 1] * w1 + EF[mrow * 3 + 2] * w2;
    Hout[mrow * HS + ng] = fmaxf(acc[r] + bn + e, 0.0f);
  }
}

// ---------------------------------------------------------------------------
// Kernel B: single workgroup (8 waves) runs the collapsed batched MLP.
//   H1 = relu(S @ W1 / 256 + b1)          (tiny, scalar: K=3)
//   H2 = relu(H1 @ W2 + b2)               (WMMA)
//   H3 = relu(H2 @ W3 + b3)               (WMMA)
//   HM = relu(H3 @ (We1a+We1b) + be1 + EF@We1e)  (WMMA)
//   out[b] = sigmoid(HM[b]·We2 + be2)  duplicated into both slots
// ---------------------------------------------------------------------------
__global__ void gnn_mlp_kernel(const float* __restrict__ ws,
                               const float* __restrict__ W1, const float* __restrict__ b1,
                               const float* __restrict__ W2, const float* __restrict__ b2,
                               const float* __restrict__ W3, const float* __restrict__ b3,
                               const float* __restrict__ We1, const float* __restrict__ be1,
                               const float* __restrict__ We2, const float* __restrict__ be2,
                               float* __restrict__ out) {
  __shared__ float bufA[BB * HS];
  __shared__ float bufB[BB * HS];
  const int tid  = threadIdx.x;
  const int lane = tid & 31;
  const int wave = tid >> 5;                 // 0..7
  const int mt   = wave >> 2;                // 0..1
  const int nt   = wave & 3;                 // 0..3
  const float* S  = ws;
  const float* EF = ws + BB * 3;

  // H1: 2048 outputs, K=3 — scalar
  for (int e = tid; e < BB * HH; e += 256) { // uniform 8 iters/thread
    const int bb = e >> 6, h = e & 63;
    const float acc = S[bb*3+0] * W1[h] + S[bb*3+1] * W1[HH + h] + S[bb*3+2] * W1[2*HH + h];
    bufA[bb * HS + h] = fmaxf(acc * (1.0f / 256.0f) + b1[h], 0.0f);
  }
  __syncthreads();

  wmma_tile_layer(bufA, bufB, W2, b2, lane, mt, nt);
  __syncthreads();
  wmma_tile_layer(bufB, bufA, W3, b3, lane, mt, nt);
  __syncthreads();
  wmma_edge_layer(bufA, bufB, We1, be1, EF, lane, mt, nt);
  __syncthreads();

  if (tid < BB) {
    float s = be2[0];
    for (int h = 0; h < HH; ++h) s += bufB[tid * HS + h] * We2[h];
    const float sig = 1.0f / (1.0f + expf(-s));
    out[2 * tid]     = sig;
    out[2 * tid + 1] = sig;   // both selected logits use ea[cur] -> identical
  }
}

extern "C" void kernel_launch(void* const* d_in, const int* in_sizes, int n_in,
                              void* d_out, int out_size, void* d_ws, size_t ws_size,
                              hipStream_t stream) {
  (void)in_sizes; (void)n_in; (void)out_size; (void)ws_size;
  const float* x   = (const float*)d_in[0];
  const float* W1  = (const float*)d_in[1];
  const float* b1  = (const float*)d_in[2];
  const float* W2  = (const float*)d_in[3];
  const float* b2  = (const float*)d_in[4];
  const float* W3  = (const float*)d_in[5];
  const float* b3  = (const float*)d_in[6];
  const float* We1 = (const float*)d_in[7];
  const float* be1 = (const float*)d_in[8];
  const float* We2 = (const float*)d_in[9];
  const float* be2 = (const float*)d_in[10];
  float* out = (float*)d_out;
  float* ws  = (float*)d_ws;   // needs 192 floats (S:96 + EF:96)

  gnn_feat_kernel<<<BB, NN, 0, stream>>>(x, ws);
  gnn_mlp_kernel<<<1, 256, 0, stream>>>(ws, W1, b1, W2, b2, W3, b3,
                                        We1, be1, We2, be2, out);
}